// BasicDeconvolutionBlock_37452114821415
// MI455X (gfx1250) — compile-verified
//
#include <hip/hip_runtime.h>
#include <hip/hip_bf16.h>

typedef __attribute__((ext_vector_type(2))) float v2f;
typedef __attribute__((ext_vector_type(8))) float v8f;

#define CIN  64
#define COUT 64
#define KOFF 27

// ---------------------------------------------------------------- zero fill
__global__ void zero_kernel(float* __restrict__ p, long n)
{
    long i = (long)blockIdx.x * blockDim.x + threadIdx.x;
    long stride = (long)gridDim.x * blockDim.x;
    for (; i < n; i += stride) p[i] = 0.0f;
}

// -------------------------------------------- gather-GEMM-scatter via WMMA
// One block = 16 rows of one offset k. 4 waves; wave w computes the
// 16(M) x 16(N) fp32 tile for output channels [16w, 16w+16) with
// 16 chained v_wmma_f32_16x16x4_f32 over K = CIN = 64.
__global__ void __launch_bounds__(128)
conv_wmma_kernel(const float* __restrict__ feats,
                 const float* __restrict__ w,
                 const int*   __restrict__ in_map,
                 const int*   __restrict__ out_map,
                 float*       __restrict__ accum,   // N x 64 (d_out)
                 int N, int emax)
{
    __shared__ float As[16][68];   // 68-stride: rows land on distinct banks
    __shared__ int   s_in[16];
    __shared__ int   s_out[16];
    __shared__ int   s_flag;

    const int  tid  = threadIdx.x;
    const int  k    = blockIdx.y;
    const long base = (long)k * emax;

    if (tid == 0) s_flag = 0;
    if (tid < 16) {
        int r = blockIdx.x * 16 + tid;
        int vin = 0, vout = N;                 // padding convention of reference
        if (r < emax) {
            vin  = in_map[base + r];
            vout = out_map[base + r];
        }
        s_in[tid]  = vin;
        s_out[tid] = vout;
    }
    __syncthreads();
    if (tid < 16 && s_out[tid] != N) s_flag = 1;
    __syncthreads();
    if (s_flag == 0) return;                   // whole tile is dummy padding

    // Gather A tile: 16 rows x 64 fp32; 128 threads x 8 floats each.
    {
        int row = tid >> 3;
        int col = (tid & 7) * 8;
        const float* src = feats + (long)s_in[row] * CIN + col;
        float4 a0 = *(const float4*)(src);
        float4 a1 = *(const float4*)(src + 4);
        *(float4*)&As[row][col]     = a0;
        *(float4*)&As[row][col + 4] = a1;
    }
    __syncthreads();

    const int wave = tid >> 5;            // 0..3  -> output-channel tile
    const int lane = tid & 31;
    const int m    = lane & 15;
    const int half = lane >> 4;           // 0: K pair {0,1}, 1: K pair {2,3}
    const int n    = (wave << 4) + m;     // output channel
    const float* W = w + (long)k * CIN * COUT;

    v8f acc = {0.f, 0.f, 0.f, 0.f, 0.f, 0.f, 0.f, 0.f};

#pragma unroll
    for (int k0 = 0; k0 < CIN; k0 += 4) {
        const int ka = k0 + 2 * half;
        v2f a, b;
        // A (16x4 fp32): lanes 0-15 VGPR0/1 = K=k0,k0+1 ; lanes 16-31 = k0+2,k0+3
        a.x = As[m][ka];
        a.y = As[m][ka + 1];
        // B (4x16 fp32): lane n holds column n; lanes 0-15 K pair {0,1}, 16-31 {2,3}
        b.x = W[(long)ka       * COUT + n];
        b.y = W[(long)(ka + 1) * COUT + n];
        acc = __builtin_amdgcn_wmma_f32_16x16x4_f32(
                  false, a, false, b, (short)0, acc, false, false);
    }

    // Scatter-add 16x16 tile. C/D layout: VGPR v, lanes 0-15 -> M=v,
    // lanes 16-31 -> M=v+8; N = lane%16 (+ wave*16).
#pragma unroll
    for (int v = 0; v < 8; ++v) {
        int row  = v + 8 * half;
        int orow = s_out[row];
        if (orow < N)                      // drop dummy rows (out_idx == N)
            atomicAdd(&accum[(long)orow * COUT + n], acc[v]);
    }
}

// ------------------------------------------------ per-channel sum / sumsq
__global__ void __launch_bounds__(256)
reduce_kernel(const float* __restrict__ accum, float* __restrict__ stats, int N)
{
    __shared__ float ssum[256];
    __shared__ float ssq[256];
    const int tid = threadIdx.x;
    const int c   = tid & 63;
    int rgrp    = (blockIdx.x * 256 + tid) >> 6;
    int rstride = (gridDim.x * 256) >> 6;
    float s = 0.f, q = 0.f;
    for (int row = rgrp; row < N; row += rstride) {
        float x = accum[(long)row * COUT + c];
        s += x;
        q += x * x;
    }
    ssum[tid] = s;
    ssq[tid]  = q;
    __syncthreads();
    if (tid < 64) {
        s = ssum[tid] + ssum[tid + 64] + ssum[tid + 128] + ssum[tid + 192];
        q = ssq[tid]  + ssq[tid + 64]  + ssq[tid + 128]  + ssq[tid + 192];
        atomicAdd(&stats[c],      s);
        atomicAdd(&stats[64 + c], q);
    }
}

// -------------------------------------------------- BN + ReLU, in place
__global__ void finalize_kernel(const float* __restrict__ stats,
                                const float* __restrict__ gamma,
                                const float* __restrict__ beta,
                                float* __restrict__ out, int N)
{
    const long total = (long)N * COUT;
    long i = (long)blockIdx.x * blockDim.x + threadIdx.x;
    long stride = (long)gridDim.x * blockDim.x;
    const float invN = 1.0f / (float)N;
    for (; i < total; i += stride) {
        int   c    = (int)(i & 63);
        float mean = stats[c] * invN;
        float var  = stats[64 + c] * invN - mean * mean;
        float x    = out[i];
        float y    = (x - mean) * rsqrtf(var + 1e-5f) * gamma[c] + beta[c];
        out[i]     = fmaxf(y, 0.0f);
    }
}

// ---------------------------------------------------------------- launcher
extern "C" void kernel_launch(void* const* d_in, const int* in_sizes, int n_in,
                              void* d_out, int out_size, void* d_ws, size_t ws_size,
                              hipStream_t stream)
{
    const float* feats   = (const float*)d_in[0];
    const float* w       = (const float*)d_in[1];
    const float* gamma   = (const float*)d_in[2];
    const float* beta    = (const float*)d_in[3];
    const int*   in_map  = (const int*)d_in[4];
    const int*   out_map = (const int*)d_in[5];

    const int N    = in_sizes[0] / CIN;
    const int emax = in_sizes[4] / KOFF;

    float* out   = (float*)d_out;     // used directly as fp32 accumulator
    float* stats = (float*)d_ws;      // 128 floats: [0..63]=sum, [64..127]=sumsq

    zero_kernel<<<1024, 256, 0, stream>>>(out, (long)N * COUT);
    zero_kernel<<<1, 128, 0, stream>>>(stats, 128);

    dim3 grid((emax + 15) / 16, KOFF);
    conv_wmma_kernel<<<grid, 128, 0, stream>>>(feats, w, in_map, out_map,
                                               out, N, emax);

    reduce_kernel<<<512, 256, 0, stream>>>(out, stats, N);

    finalize_kernel<<<1024, 256, 0, stream>>>(stats, gamma, beta, out, N);
}